// EuclideanLossWithOHEM_57818849738795
// MI455X (gfx1250) — compile-verified
//
#include <hip/hip_runtime.h>

// EuclideanLossWithOHEM for MI455X (gfx1250, wave32).
// Bandwidth-bound (~100MB => ~5us at 23.3TB/s). One streaming main pass +
// exact top-k threshold via 3-level radix histogram on float bits.
// CDNA5 paths: v_wmma_f32_16x16x4_f32 cross-lane reductions,
// global_load_async_to_lds_b128 tile staging (ASYNCcnt), NT load hints.

#define NIMG 16
#define HW   (512 * 512)

typedef __attribute__((ext_vector_type(2))) float v2f;
typedef __attribute__((ext_vector_type(4))) float v4f;
typedef __attribute__((ext_vector_type(8))) float v8f;
typedef __attribute__((ext_vector_type(4))) int   v4i;

// Full-wave (32 lane) sum using the matrix unit:
// A = ones (16x4), B vgpr0 = v, vgpr1 = 0  =>  D[m][n] = v_n + v_{n+16}.
// D vgpr0 lane L holds colsum[L & 15]; 4 xor-shuffles finish the 16-way tree.
__device__ __forceinline__ float wave_reduce_sum(float v) {
  v2f a; a[0] = 1.0f; a[1] = 1.0f;
  v2f b; b[0] = v;    b[1] = 0.0f;
  v8f c = {};
  c = __builtin_amdgcn_wmma_f32_16x16x4_f32(false, a, false, b, (short)0, c,
                                            false, false);
  float s = c[0];
  s += __shfl_xor(s, 1, 32);
  s += __shfl_xor(s, 2, 32);
  s += __shfl_xor(s, 4, 32);
  s += __shfl_xor(s, 8, 32);
  return s;
}

// ---- async global->LDS (gfx1250 ASYNCcnt path) via inline asm ----
// VFLAT form: global_load_async_to_lds_b128 vdst(LDS byte addr), vaddr, off
// Flat pointers to __shared__ have LDS offset in addr[31:0],
// so VDST = low 32 bits of the flat ptr.
__device__ __forceinline__ void async_copy16(const float* g, float* l) {
  unsigned lds_addr = (unsigned)(uintptr_t)l;
  asm volatile("global_load_async_to_lds_b128 %0, %1, off"
               :
               : "v"(lds_addr), "v"(g)
               : "memory");
}

__device__ __forceinline__ void wait_async0() {
#if __has_builtin(__builtin_amdgcn_s_wait_asynccnt)
  __builtin_amdgcn_s_wait_asynccnt(0);
#else
  asm volatile("s_wait_asynccnt 0x0" ::: "memory");
#endif
}

// ---------------- workspace zeroing ----------------
__global__ void k_zero(unsigned* ws, int words) {
  int i = blockIdx.x * blockDim.x + threadIdx.x;
  if (i < words) ws[i] = 0u;
}

// ---------------- per-image label counts ----------------
// grid = 1024 (64 blocks/image), block = 256, 16 px/thread via int4
__global__ void k_label_count(const int* __restrict__ gt,
                              unsigned* __restrict__ counts) {
  __shared__ unsigned lc[16];
  if (threadIdx.x < 16) lc[threadIdx.x] = 0u;
  __syncthreads();
  const int img = blockIdx.x >> 6;
  const size_t base = (size_t)img * HW + (size_t)(blockIdx.x & 63) * 4096;
#pragma unroll
  for (int i = 0; i < 4; ++i) {
    v4i g4 = __builtin_nontemporal_load(
        (const v4i*)(gt + base + ((size_t)threadIdx.x + i * 256) * 4));
    atomicAdd(&lc[g4.x & 15], 1u);
    atomicAdd(&lc[g4.y & 15], 1u);
    atomicAdd(&lc[g4.z & 15], 1u);
    atomicAdd(&lc[g4.w & 15], 1u);
  }
  __syncthreads();
  if (threadIdx.x < 16 && lc[threadIdx.x])
    atomicAdd(&counts[img * 16 + threadIdx.x], lc[threadIdx.x]);
}

// ---------------- per-image label weight table + hard-negative k ----------
__global__ void k_seg(const unsigned* __restrict__ counts,
                      int* __restrict__ kHard, float* __restrict__ wTab) {
  int n = threadIdx.x;
  if (n < NIMG) {
    unsigned c[9];
    unsigned pos = 0; int seg = 0;
    for (int l = 1; l < 9; ++l) {
      c[l] = counts[n * 16 + l];
      pos += c[l]; seg += (c[l] > 0u) ? 1 : 0;
    }
    float sa = (seg > 0) ? (float)pos / (float)seg : 0.0f;
    wTab[n * 16 + 0] = 0.0f;
    for (int l = 1; l < 9; ++l)
      wTab[n * 16 + l] = (c[l] > 0u) ? sa / (float)c[l] : 0.0f;
    for (int l = 9; l < 16; ++l) wTab[n * 16 + l] = 0.0f;
    int sumNeg = HW - (int)pos;
    int k3 = 3 * (int)pos;
    kHard[n] = (k3 < sumNeg) ? k3 : sumNeg;
  }
}

// ---------------- main streaming pass ----------------
// grid = 4096 (256 blocks/image), block = 256, 4 contiguous px/thread (b128)
__global__ void k_prep(const float* __restrict__ pred,
                       const float* __restrict__ gt_df,
                       const int* __restrict__ gt,
                       const float* __restrict__ wTab,
                       float* __restrict__ lossNeg,
                       float* __restrict__ sumW, float* __restrict__ numPos,
                       unsigned* __restrict__ h1c, float* __restrict__ h1s) {
  __shared__ unsigned s_cnt[1024];
  __shared__ float    s_sum[1024];
  __shared__ float    s_w[16];
  __shared__ float    s_red[8][2];
  const int img = blockIdx.x >> 8;
  if (threadIdx.x < 16) s_w[threadIdx.x] = wTab[img * 16 + threadIdx.x];
  for (int i = threadIdx.x; i < 1024; i += 256) { s_cnt[i] = 0u; s_sum[i] = 0.f; }
  __syncthreads();

  const size_t p = (size_t)(blockIdx.x & 255) * 1024 + (size_t)threadIdx.x * 4;
  const size_t imgoff = (size_t)img * HW;
  const size_t c0 = (size_t)img * 2 * HW;

  v4f a0 = __builtin_nontemporal_load((const v4f*)(pred  + c0 + p));
  v4f b0 = __builtin_nontemporal_load((const v4f*)(gt_df + c0 + p));
  v4f a1 = __builtin_nontemporal_load((const v4f*)(pred  + c0 + HW + p));
  v4f b1 = __builtin_nontemporal_load((const v4f*)(gt_df + c0 + HW + p));
  v4i g4 = __builtin_nontemporal_load((const v4i*)(gt + imgoff + p));

  v4f d0 = a0 - b0;
  v4f d1 = a1 - b1;
  v4f l2 = d0 * d0 + d1 * d1;
  v4f ln;
  float myW = 0.f, myNum = 0.f;
#pragma unroll
  for (int e = 0; e < 4; ++e) {
    float v = l2[e];
    int   g = g4[e];
    if (g > 0) {
      float w = s_w[g & 15];           // no divide in the hot loop
      myW += w;
      myNum += v * w;
      ln[e] = -1.0f;                   // sentinel: positive pixel (sign bit set)
    } else {
      ln[e] = v;
      unsigned bits = __float_as_uint(v);
      atomicAdd(&s_cnt[bits >> 21], 1u);   // top 10 bits (sign is 0)
      atomicAdd(&s_sum[bits >> 21], v);
    }
  }
  *(v4f*)(lossNeg + imgoff + p) = ln;

  // block reduction: WMMA per-wave, LDS across the 8 waves
  float rW = wave_reduce_sum(myW);
  float rN = wave_reduce_sum(myNum);
  int wid = threadIdx.x >> 5;
  if ((threadIdx.x & 31) == 0) { s_red[wid][0] = rW; s_red[wid][1] = rN; }
  __syncthreads();
  if (threadIdx.x == 0) {
    float tW = 0.f, tN = 0.f;
    for (int w = 0; w < 8; ++w) { tW += s_red[w][0]; tN += s_red[w][1]; }
    atomicAdd(&sumW[img], tW);
    atomicAdd(&numPos[img], tN);
  }
  __syncthreads();
  for (int i = threadIdx.x; i < 1024; i += 256) {
    if (s_cnt[i]) {
      atomicAdd(&h1c[img * 1024 + i], s_cnt[i]);
      atomicAdd(&h1s[img * 1024 + i], s_sum[i]);
    }
  }
}

// ---------------- level-1 select ----------------
__global__ void k_select1(const unsigned* __restrict__ h1c,
                          const float* __restrict__ h1s,
                          const int* __restrict__ kHard,
                          int* bin1, int* cntA1, float* sumA1) {
  __shared__ unsigned c[1024];
  __shared__ float    s[1024];
  const int img = blockIdx.x;
  c[threadIdx.x] = h1c[img * 1024 + threadIdx.x];
  s[threadIdx.x] = h1s[img * 1024 + threadIdx.x];
  __syncthreads();
  if (threadIdx.x == 0) {
    int k = kHard[img];
    int b = -1, cA = 0; float sA = 0.f;
    if (k > 0) {
      int cum = 0; float sum = 0.f;
      for (b = 1023; b >= 0; --b) {
        int nc = cum + (int)c[b];
        if (nc >= k) { cA = cum; sA = sum; break; }
        cum = nc; sum += s[b];
      }
    }
    bin1[img] = b; cntA1[img] = cA; sumA1[img] = sA;
  }
}

// ---------------- level-2 histogram (bits[20:10]) ----------------
// grid = NIMG*128, block = 256; 8KB tile staged via async-to-LDS
__global__ void k_hist2(const float* __restrict__ lossNeg,
                        const int* __restrict__ bin1,
                        unsigned* __restrict__ h2c, float* __restrict__ h2s) {
  const int img = blockIdx.x >> 7;
  const int b1 = bin1[img];
  if (b1 < 0) return;
  __shared__ float    tile[2048];
  __shared__ unsigned sc[2048];
  __shared__ float    ss[2048];
  const size_t base = (size_t)img * HW + (size_t)(blockIdx.x & 127) * 2048;
  async_copy16(lossNeg + base + (size_t)threadIdx.x * 4,
               &tile[threadIdx.x * 4]);
  async_copy16(lossNeg + base + 1024 + (size_t)threadIdx.x * 4,
               &tile[1024 + threadIdx.x * 4]);
  for (int i = threadIdx.x; i < 2048; i += 256) { sc[i] = 0u; ss[i] = 0.f; }
  wait_async0();
  __syncthreads();
  for (int i = threadIdx.x; i < 2048; i += 256) {
    float v = tile[i];
    unsigned bits = __float_as_uint(v);
    if ((int)(bits >> 21) == b1) {
      unsigned b = (bits >> 10) & 2047u;
      atomicAdd(&sc[b], 1u);
      atomicAdd(&ss[b], v);
    }
  }
  __syncthreads();
  for (int i = threadIdx.x; i < 2048; i += 256) {
    if (sc[i]) {
      atomicAdd(&h2c[img * 2048 + i], sc[i]);
      atomicAdd(&h2s[img * 2048 + i], ss[i]);
    }
  }
}

// ---------------- level-2 select ----------------
__global__ void k_select2(const unsigned* __restrict__ h2c,
                          const float* __restrict__ h2s,
                          const int* __restrict__ kHard,
                          const int* __restrict__ bin1,
                          const int* __restrict__ cntA1,
                          const float* __restrict__ sumA1,
                          int* bin2, int* cntA2, float* sumA2) {
  __shared__ unsigned c[2048];
  __shared__ float    s[2048];
  const int img = blockIdx.x;
  c[threadIdx.x]        = h2c[img * 2048 + threadIdx.x];
  c[threadIdx.x + 1024] = h2c[img * 2048 + 1024 + threadIdx.x];
  s[threadIdx.x]        = h2s[img * 2048 + threadIdx.x];
  s[threadIdx.x + 1024] = h2s[img * 2048 + 1024 + threadIdx.x];
  __syncthreads();
  if (threadIdx.x == 0) {
    int k = kHard[img];
    int b = -1, cA = 0; float sA = 0.f;
    if (k > 0 && bin1[img] >= 0) {
      int cum = cntA1[img]; float sum = sumA1[img];
      for (b = 2047; b >= 0; --b) {
        int nc = cum + (int)c[b];
        if (nc >= k) { cA = cum; sA = sum; break; }
        cum = nc; sum += s[b];
      }
    }
    bin2[img] = b; cntA2[img] = cA; sumA2[img] = sA;
  }
}

// ---------------- level-3 histogram (bits[9:0]) ----------------
__global__ void k_hist3(const float* __restrict__ lossNeg,
                        const int* __restrict__ bin1,
                        const int* __restrict__ bin2,
                        unsigned* __restrict__ h3c, float* __restrict__ h3s) {
  const int img = blockIdx.x >> 7;
  const int b1 = bin1[img];
  if (b1 < 0) return;
  const unsigned hi = ((unsigned)b1 << 11) | (unsigned)bin2[img];
  __shared__ float    tile[2048];
  __shared__ unsigned sc[1024];
  __shared__ float    ss[1024];
  const size_t base = (size_t)img * HW + (size_t)(blockIdx.x & 127) * 2048;
  async_copy16(lossNeg + base + (size_t)threadIdx.x * 4,
               &tile[threadIdx.x * 4]);
  async_copy16(lossNeg + base + 1024 + (size_t)threadIdx.x * 4,
               &tile[1024 + threadIdx.x * 4]);
  for (int i = threadIdx.x; i < 1024; i += 256) { sc[i] = 0u; ss[i] = 0.f; }
  wait_async0();
  __syncthreads();
  for (int i = threadIdx.x; i < 2048; i += 256) {
    float v = tile[i];
    unsigned bits = __float_as_uint(v);
    if ((bits >> 10) == hi) {
      unsigned b = bits & 1023u;
      atomicAdd(&sc[b], 1u);
      atomicAdd(&ss[b], v);
    }
  }
  __syncthreads();
  for (int i = threadIdx.x; i < 1024; i += 256) {
    if (sc[i]) {
      atomicAdd(&h3c[img * 1024 + i], sc[i]);
      atomicAdd(&h3s[img * 1024 + i], ss[i]);
    }
  }
}

// ---------------- level-3 select: exact threshold ----------------
__global__ void k_select3(const unsigned* __restrict__ h3c,
                          const float* __restrict__ h3s,
                          const int* __restrict__ kHard,
                          const int* __restrict__ bin1,
                          const int* __restrict__ bin2,
                          const int* __restrict__ cntA2,
                          const float* __restrict__ sumA2,
                          float* Tval, int* cntG, float* sumG) {
  __shared__ unsigned c[1024];
  __shared__ float    s[1024];
  const int img = blockIdx.x;
  c[threadIdx.x] = h3c[img * 1024 + threadIdx.x];
  s[threadIdx.x] = h3s[img * 1024 + threadIdx.x];
  __syncthreads();
  if (threadIdx.x == 0) {
    int k = kHard[img];
    float T = 0.f; int cG = 0; float sG = 0.f;
    if (k > 0 && bin1[img] >= 0 && bin2[img] >= 0) {
      int cum = cntA2[img]; float sum = sumA2[img];
      int b;
      for (b = 1023; b >= 0; --b) {
        int nc = cum + (int)c[b];
        if (nc >= k) { cG = cum; sG = sum; break; }
        cum = nc; sum += s[b];
      }
      unsigned tb = ((unsigned)bin1[img] << 21) |
                    ((unsigned)bin2[img] << 10) | (unsigned)b;
      T = __uint_as_float(tb);
    }
    Tval[img] = T; cntG[img] = cG; sumG[img] = sG;
  }
}

// ---------------- final scalar ----------------
__global__ void k_final(const int* __restrict__ kHard,
                        const float* __restrict__ Tval,
                        const int* __restrict__ cntG,
                        const float* __restrict__ sumG,
                        const float* __restrict__ numPos,
                        const float* __restrict__ sumW,
                        float* __restrict__ out) {
  const int n = threadIdx.x;
  float num = 0.f, den = 0.f;
  if (n < NIMG) {
    int k = kHard[n];
    float negNum = 0.f, negW = 0.f;
    if (k > 0) {
      float T = Tval[n]; int cg = cntG[n]; float sg = sumG[n];
      if (T > 0.f) { negW = (float)k; negNum = sg + (float)(k - cg) * T; }
      else         { negW = (float)cg; negNum = sg; }
    }
    num = numPos[n] + negNum;
    den = sumW[n] + negW;
  }
  float tn = wave_reduce_sum(num);
  float td = wave_reduce_sum(den);
  if (n == 0) out[0] = tn / (64.0f * td);   // /N /2 /(2*sum_wTot)
}

// ---------------- host launch ----------------
extern "C" void kernel_launch(void* const* d_in, const int* in_sizes, int n_in,
                              void* d_out, int out_size, void* d_ws,
                              size_t ws_size, hipStream_t stream) {
  (void)in_sizes; (void)n_in; (void)out_size; (void)ws_size;
  const float* pred  = (const float*)d_in[0];
  const float* gt_df = (const float*)d_in[1];
  const int*   gt    = (const int*)d_in[2];
  float* out = (float*)d_out;
  char* ws = (char*)d_ws;

  unsigned* counts = (unsigned*)(ws + 0);        // NIMG*16 u32
  int*      kHard  = (int*)(ws + 4160);
  float*    sumW   = (float*)(ws + 4224);
  float*    numPos = (float*)(ws + 4288);
  int*      bin1   = (int*)(ws + 4352);
  int*      cntA1  = (int*)(ws + 4416);
  float*    sumA1  = (float*)(ws + 4480);
  int*      bin2   = (int*)(ws + 4544);
  int*      cntA2  = (int*)(ws + 4608);
  float*    sumA2  = (float*)(ws + 4672);
  float*    Tval   = (float*)(ws + 4736);
  int*      cntG   = (int*)(ws + 4800);
  float*    sumG   = (float*)(ws + 4864);
  float*    wTab   = (float*)(ws + 5120);        // NIMG*16 f32
  unsigned* h1c = (unsigned*)(ws + 8192);        // NIMG*1024 u32
  float*    h1s = (float*)(ws + 73728);          // NIMG*1024 f32
  unsigned* h2c = (unsigned*)(ws + 139264);      // NIMG*2048 u32
  float*    h2s = (float*)(ws + 270336);         // NIMG*2048 f32
  unsigned* h3c = (unsigned*)(ws + 401408);      // NIMG*1024 u32
  float*    h3s = (float*)(ws + 465920);         // NIMG*1024 f32
  float*    lossNeg = (float*)(ws + (1u << 20)); // NIMG*HW f32 (16 MB)

  const int zwords = 531456 / 4;                 // zero accum region each call
  k_zero<<<(zwords + 1023) / 1024, 1024, 0, stream>>>((unsigned*)ws, zwords);
  k_label_count<<<1024, 256, 0, stream>>>(gt, counts);
  k_seg<<<1, 32, 0, stream>>>(counts, kHard, wTab);
  k_prep<<<4096, 256, 0, stream>>>(pred, gt_df, gt, wTab,
                                   lossNeg, sumW, numPos, h1c, h1s);
  k_select1<<<NIMG, 1024, 0, stream>>>(h1c, h1s, kHard, bin1, cntA1, sumA1);
  k_hist2<<<NIMG * 128, 256, 0, stream>>>(lossNeg, bin1, h2c, h2s);
  k_select2<<<NIMG, 1024, 0, stream>>>(h2c, h2s, kHard, bin1, cntA1, sumA1,
                                       bin2, cntA2, sumA2);
  k_hist3<<<NIMG * 128, 256, 0, stream>>>(lossNeg, bin1, bin2, h3c, h3s);
  k_select3<<<NIMG, 1024, 0, stream>>>(h3c, h3s, kHard, bin1, bin2, cntA2,
                                       sumA2, Tval, cntG, sumG);
  k_final<<<1, 32, 0, stream>>>(kHard, Tval, cntG, sumG, numPos, sumW, out);
}